// SpatialSelfAttention_85968065397288
// MI455X (gfx1250) — compile-verified
//
#include <hip/hip_runtime.h>

// Problem constants (from the reference)
#define BS_     4
#define NQ_     6400
#define DIMS_   256
#define NH_     8
#define NP_     4
#define NQUEUE_ 2
#define NRAW_   192   // 128 offsets + 64 attn logits per row

typedef __attribute__((ext_vector_type(2))) float v2f;
typedef __attribute__((ext_vector_type(8))) float v8f;

// ---------------------------------------------------------------------------
// WMMA f32 GEMM:  out[M,NDIM] = A[M,KDIM] @ W[KDIM,NDIM] + bias (+ resid)
// MODE 0: A = interleaved value rows (query/voxbev), W=Wv, bias=bv
// MODE 1: A = qcat rows (value[:4] || query), W = [Woff | Wattn], bias = [boff|battn]
// MODE 2: A = sampled output, W=Wo, bias=bo, + residual (query)
//
// Block: 256 threads = 8 waves; wave computes 16 rows x NDIM cols.
// Block tile: 128 rows; K staged in chunks of 16 through LDS.
// LDS tiles are stored PAIR-INTERLEAVED ([kpair][idx][2]) so that each WMMA
// fragment (A: {A[m][k],A[m][k+1]}, B: {W[k][n],W[k+1][n]}) is one aligned
// ds_load_b64 — no v_mov re-packing. Plane strides are padded so the two
// half-waves (k-plane p and p+1) hit disjoint 32-bank sets (stride ≡ 32 mod 64).
// Global->LDS staging is software-pipelined: chunk k+1 global loads issue
// under the 64-WMMA burst of chunk k.
// ---------------------------------------------------------------------------
template<int KDIM, int NDIM, int MODE>
__global__ __launch_bounds__(256)
void gemm_wmma_kernel(const float* __restrict__ Aq, const float* __restrict__ Av,
                      const float* __restrict__ W0, const float* __restrict__ W1,
                      const float* __restrict__ B0, const float* __restrict__ B1,
                      const float* __restrict__ resid, float* __restrict__ out)
{
  constexpr int NT    = NDIM / 16;   // 16 or 12 accumulators per wave
  constexpr int WITER = NDIM / 32;   // W pairs loaded per thread per chunk

  // sA plane stride = 144*2 = 288 dwords ≡ 32 (mod 64); rows 128..143 are pad.
  __shared__ float sA[8][144][2];
  // sW plane stride = (NDIM+16)*2 dwords ≡ 32 (mod 64).
  __shared__ float sW[8][NDIM + 16][2];

  const int t    = threadIdx.x;
  const int wave = t >> 5;
  const int lane = t & 31;
  const int lm   = lane & 15;        // M (A frag) / N (B,C,D frag) index
  const int kh   = lane >> 4;        // half-wave: selects k-plane p / p+1
  const long long rowBase = (long long)blockIdx.x * 128;

  v8f zero = {};
  v8f acc[NT];
#pragma unroll
  for (int i = 0; i < NT; ++i) acc[i] = zero;

  // A staging: thread t <-> row (t & 127), 8-float chunk ((t>>7)*8).
  // Conflict-free LDS stores: within a wave, banks = 2*ti, 2*ti+1.
  const int ar = t & 127;
  const int ac = (t >> 7) * 8;
  const long long g    = rowBase + ar;
  const long long bIdx = g / NQ_;
  const long long pos  = g - bIdx * NQ_;

  auto loadA = [&](int k0, float* areg) {
    const float* src;
    if constexpr (MODE == 0) {
      // value[vb] = (vb odd) ? voxbev[vb>>1] : query[vb>>1]
      const float* base = (bIdx & 1) ? Av : Aq;
      src = base + ((bIdx >> 1) * NQ_ + pos) * DIMS_ + (k0 + ac);
    } else if constexpr (MODE == 1) {
      const int c = k0 + ac;                     // 8-aligned: uniform side of 256
      if (c < DIMS_) {
        const float* base = (bIdx & 1) ? Av : Aq; // qcat low half = value[:4]
        src = base + ((bIdx >> 1) * NQ_ + pos) * DIMS_ + c;
      } else {
        src = Aq + (bIdx * NQ_ + pos) * DIMS_ + (c - DIMS_); // high half = query
      }
    } else {
      src = Aq + g * (long long)KDIM + (k0 + ac);
    }
#pragma unroll
    for (int j = 0; j < 8; ++j) areg[j] = src[j];
  };

  auto loadW = [&](int k0, v2f* wreg) {
#pragma unroll
    for (int i = 0; i < WITER; ++i) {
      const int idx = t + i * 256;     // over 8*NDIM k-pair x col slots
      const int kp  = idx / NDIM;
      const int n   = idx - kp * NDIM;
      const int row = k0 + 2 * kp;
      float w0, w1;
      if constexpr (MODE == 1) {
        if (n < 128) {
          w0 = W0[(long long)row * 128 + n];
          w1 = W0[(long long)(row + 1) * 128 + n];
        } else {
          w0 = W1[(long long)row * 64 + (n - 128)];
          w1 = W1[(long long)(row + 1) * 64 + (n - 128)];
        }
      } else {
        w0 = W0[(long long)row * NDIM + n];
        w1 = W0[(long long)(row + 1) * NDIM + n];
      }
      v2f wv; wv.x = w0; wv.y = w1;
      wreg[i] = wv;
    }
  };

  auto storeLDS = [&](const float* areg, const v2f* wreg) {
#pragma unroll
    for (int jj = 0; jj < 4; ++jj) {
      v2f av; av.x = areg[2 * jj]; av.y = areg[2 * jj + 1];
      *reinterpret_cast<v2f*>(&sA[(ac >> 1) + jj][ar][0]) = av;
    }
#pragma unroll
    for (int i = 0; i < WITER; ++i) {
      const int idx = t + i * 256;
      const int kp  = idx / NDIM;
      const int n   = idx - kp * NDIM;
      *reinterpret_cast<v2f*>(&sW[kp][n][0]) = wreg[i];
    }
  };

  float areg[8];
  v2f   wreg[WITER];
  loadA(0, areg);
  loadW(0, wreg);

  for (int k0 = 0; k0 < KDIM; k0 += 16) {
    storeLDS(areg, wreg);
    __syncthreads();
    if (k0 + 16 < KDIM) {              // prefetch next chunk under the WMMAs
      loadA(k0 + 16, areg);
      loadW(k0 + 16, wreg);
    }
#pragma unroll
    for (int kk = 0; kk < 16; kk += 4) {
      const int kp = (kk >> 1) + kh;   // §7.12.2: K = 2*(lane>=16) + j
      const v2f a = *reinterpret_cast<const v2f*>(&sA[kp][wave * 16 + lm][0]);
#pragma unroll
      for (int nt = 0; nt < NT; ++nt) {
        const v2f b = *reinterpret_cast<const v2f*>(&sW[kp][nt * 16 + lm][0]);
        acc[nt] = __builtin_amdgcn_wmma_f32_16x16x4_f32(
            false, a, false, b, (short)0, acc[nt], false, false);
      }
    }
    __syncthreads();
  }

  // ---- epilogue: C/D layout M = 8*kh + r, N = lm ----
#pragma unroll
  for (int nt = 0; nt < NT; ++nt) {
#pragma unroll
    for (int r = 0; r < 8; ++r) {
      const long long row = rowBase + wave * 16 + kh * 8 + r;
      const int col = nt * 16 + lm;
      float bias;
      if constexpr (MODE == 1) bias = (col < 128) ? B0[col] : B1[col - 128];
      else                     bias = B0[col];
      float val = acc[nt][r] + bias;
      if constexpr (MODE == 2) val += resid[row * NDIM + col];
      out[row * NDIM + col] = val;
    }
  }
}

// ---------------------------------------------------------------------------
// Softmax over P=4 + precompute pixel coords:  px = ref_x*W + off_x - 0.5
// One thread per (b, q, h, queue). sinfo[vb,q,h,p] = {px, py, weight}
// ---------------------------------------------------------------------------
__global__ __launch_bounds__(256)
void loc_attn_kernel(const float* __restrict__ raw, const float* __restrict__ ref,
                     const long long* __restrict__ shapes, float* __restrict__ sinfo)
{
  const int idx = blockIdx.x * 256 + threadIdx.x;
  if (idx >= BS_ * NQ_ * NH_ * NQUEUE_) return;
  const int queue = idx & 1;
  const int h     = (idx >> 1) & (NH_ - 1);
  const int rest  = idx >> 4;             // b*NQ + q
  const int q     = rest % NQ_;
  const int b     = rest / NQ_;

  const float Hl = (float)shapes[0];
  const float Wl = (float)shapes[1];
  const float* r = raw + (size_t)rest * NRAW_;
  const int hq = h * NQUEUE_ + queue;

  float av[4];
#pragma unroll
  for (int p = 0; p < 4; ++p) av[p] = r[128 + hq * 4 + p];
  float m = fmaxf(fmaxf(av[0], av[1]), fmaxf(av[2], av[3]));
  float e[4], s = 0.f;
#pragma unroll
  for (int p = 0; p < 4; ++p) { e[p] = __expf(av[p] - m); s += e[p]; }
  const float inv = 1.0f / s;

  const int vb = b * NQUEUE_ + queue;
  const float rx = ref[((size_t)vb * NQ_ + q) * 2 + 0];
  const float ry = ref[((size_t)vb * NQ_ + q) * 2 + 1];
  float* o = sinfo + ((((size_t)vb * NQ_ + q) * NH_ + h) * NP_) * 3;
#pragma unroll
  for (int p = 0; p < 4; ++p) {
    const float ox = r[(hq * 4 + p) * 2 + 0];
    const float oy = r[(hq * 4 + p) * 2 + 1];
    o[p * 3 + 0] = rx * Wl + ox - 0.5f;   // loc_x*W - 0.5 with off/W folded in
    o[p * 3 + 1] = ry * Hl + oy - 0.5f;
    o[p * 3 + 2] = e[p] * inv;
  }
}

// ---------------------------------------------------------------------------
// Deformable bilinear sampling + mean over queue.
// One wave per (b, q, h); lane = channel d in 0..31 -> 128B coalesced gathers.
// v is 52MB and L2 is 192MB, so the gathers are L2-resident.
// ---------------------------------------------------------------------------
__global__ __launch_bounds__(256)
void sample_kernel(const float* __restrict__ v, const float* __restrict__ sinfo,
                   const long long* __restrict__ shapes, float* __restrict__ outp)
{
  const int wid  = blockIdx.x * 8 + (threadIdx.x >> 5);
  const int lane = threadIdx.x & 31;
  const int h    = wid % NH_;
  const int rest = wid / NH_;
  const int q    = rest % NQ_;
  const int b    = rest / NQ_;
  const int Hl = (int)shapes[0];
  const int Wl = (int)shapes[1];

  float acc = 0.f;
#pragma unroll
  for (int s = 0; s < NQUEUE_; ++s) {
    const int vb = b * NQUEUE_ + s;
    const float* vp  = v + ((size_t)vb * NQ_) * DIMS_ + h * 32 + lane;
    const float* inf = sinfo + ((((size_t)vb * NQ_ + q) * NH_ + h) * NP_) * 3;
#pragma unroll
    for (int p = 0; p < NP_; ++p) {
      const float px = inf[p * 3 + 0];
      const float py = inf[p * 3 + 1];
      const float w  = inf[p * 3 + 2];
      const float fx = floorf(px), fy = floorf(py);
      const float dx = px - fx,  dy = py - fy;
      const int x0 = (int)fx, y0 = (int)fy;
      const float w00 = (1.f - dx) * (1.f - dy);
      const float w10 = dx * (1.f - dy);
      const float w01 = (1.f - dx) * dy;
      const float w11 = dx * dy;
      if (x0 >= 0     && x0 < Wl     && y0 >= 0     && y0 < Hl)
        acc += w * w00 * vp[(size_t)(y0 * Wl + x0) * DIMS_];
      if (x0 + 1 >= 0 && x0 + 1 < Wl && y0 >= 0     && y0 < Hl)
        acc += w * w10 * vp[(size_t)(y0 * Wl + x0 + 1) * DIMS_];
      if (x0 >= 0     && x0 < Wl     && y0 + 1 >= 0 && y0 + 1 < Hl)
        acc += w * w01 * vp[(size_t)((y0 + 1) * Wl + x0) * DIMS_];
      if (x0 + 1 >= 0 && x0 + 1 < Wl && y0 + 1 >= 0 && y0 + 1 < Hl)
        acc += w * w11 * vp[(size_t)((y0 + 1) * Wl + x0 + 1) * DIMS_];
    }
  }
  outp[((size_t)(b * NQ_ + q)) * DIMS_ + h * 32 + lane] = 0.5f * acc;  // mean over queue
}

// ---------------------------------------------------------------------------
extern "C" void kernel_launch(void* const* d_in, const int* in_sizes, int n_in,
                              void* d_out, int out_size, void* d_ws, size_t ws_size,
                              hipStream_t stream) {
  (void)in_sizes; (void)n_in; (void)out_size; (void)ws_size;
  const float* query  = (const float*)d_in[0];
  const float* voxbev = (const float*)d_in[1];
  const float* refp   = (const float*)d_in[2];
  const long long* shapes = (const long long*)d_in[3];
  // d_in[4] = level_start_index (single level, 0) -> unused
  const float* Wv    = (const float*)d_in[5];
  const float* bv    = (const float*)d_in[6];
  const float* Woff  = (const float*)d_in[7];
  const float* boff  = (const float*)d_in[8];
  const float* Wattn = (const float*)d_in[9];
  const float* battn = (const float*)d_in[10];
  const float* Wo    = (const float*)d_in[11];
  const float* bo    = (const float*)d_in[12];
  float* out = (float*)d_out;

  // Workspace layout (floats):
  //   vws   : 8*NQ*256   = 13,107,200  (projected values)
  //   sinfo : 8*NQ*8*4*3 =  4,915,200  (px, py, weight)
  //   shared: max(raw 25600*192, samp 25600*256) = 6,553,600
  // raw (GEMM1 output) and samp (sampler output) alias: raw's last reader
  // (loc_attn) runs before samp's writer (sample) on the same stream.
  float* ws     = (float*)d_ws;
  float* vws    = ws;
  float* sinfo  = vws + (size_t)8 * NQ_ * DIMS_;
  float* shared = sinfo + (size_t)8 * NQ_ * NH_ * NP_ * 3;
  float* raw    = shared;
  float* samp   = shared;

  // 1) value projection: [51200,256] @ Wv[256,256] + bv
  gemm_wmma_kernel<256, 256, 0><<<(8 * NQ_) / 128, 256, 0, stream>>>(
      query, voxbev, Wv, nullptr, bv, nullptr, nullptr, vws);

  // 2) offsets + attn logits: qcat[25600,512] @ [Woff|Wattn][512,192] + [boff|battn]
  gemm_wmma_kernel<512, NRAW_, 1><<<(BS_ * NQ_) / 128, 256, 0, stream>>>(
      query, voxbev, Woff, Wattn, boff, battn, nullptr, raw);

  // 3) softmax + sampling coordinates
  loc_attn_kernel<<<(BS_ * NQ_ * NH_ * NQUEUE_) / 256, 256, 0, stream>>>(
      raw, refp, shapes, sinfo);

  // 4) deformable bilinear gather + queue mean
  sample_kernel<<<(BS_ * NQ_ * NH_) / 8, 256, 0, stream>>>(
      vws, sinfo, shapes, samp);

  // 5) output projection + bias + residual
  gemm_wmma_kernel<256, 256, 2><<<(BS_ * NQ_) / 128, 256, 0, stream>>>(
      samp, nullptr, Wo, nullptr, bo, nullptr, query, out);
}